// LCSMatcher_59373627900172
// MI455X (gfx1250) — compile-verified
//
#include <hip/hip_runtime.h>
#include <hip/hip_bf16.h>

#define BS 8
#define Q 1000
#define NC 92
#define T 300
#define QP 1008   // Q padded to multiple of 16
#define CP 96     // NC padded to multiple of 4 (WMMA K)
#define TP 304    // T padded to multiple of 16
#define SROW 301  // T+1
#define SCOL 1001 // Q+1

typedef __attribute__((ext_vector_type(2))) float v2f;
typedef __attribute__((ext_vector_type(8))) float v8f;

// ---------------- softmax over classes (padded rows/cols set to 0) --------
__global__ void lcs_softmax_kernel(const float* __restrict__ logits,
                                   float* __restrict__ probs) {
  int row = blockIdx.x;                  // 0 .. BS*QP-1
  int b = row / QP, q = row % QP;
  int c = threadIdx.x;                   // 0..127
  __shared__ float red[128];
  bool act = (q < Q) && (c < NC);
  float x = act ? logits[((size_t)b * Q + q) * NC + c] : -3.0e38f;
  red[c] = x;
  __syncthreads();
  for (int s = 64; s > 0; s >>= 1) {
    if (c < s) red[c] = fmaxf(red[c], red[c + s]);
    __syncthreads();
  }
  float m = red[0];
  __syncthreads();
  float e = act ? expf(x - m) : 0.f;
  red[c] = e;
  __syncthreads();
  for (int s = 64; s > 0; s >>= 1) {
    if (c < s) red[c] += red[c + s];
    __syncthreads();
  }
  float sum = red[0];
  if (c < CP) probs[(size_t)row * CP + c] = act ? (e / sum) : 0.f;
}

// ---------------- build one-hot B matrix [BS][CP][TP] ---------------------
__global__ void lcs_onehot_kernel(const int* __restrict__ labels,
                                  float* __restrict__ oh) {
  int idx = blockIdx.x * blockDim.x + threadIdx.x;
  if (idx >= BS * CP * TP) return;
  int n = idx % TP;
  int k = (idx / TP) % CP;
  int b = idx / (TP * CP);
  float v = 0.f;
  if (n < T) v = (labels[b * T + n] == k) ? 1.f : 0.f;
  oh[idx] = v;
}

// ---------------- fused cost: WMMA class-prob GEMM + L1 + GIoU ------------
__global__ void lcs_cost_wmma_kernel(const float* __restrict__ probs,
                                     const float* __restrict__ oh,
                                     const float* __restrict__ pboxes,
                                     const float* __restrict__ tboxes,
                                     float* __restrict__ cost) {
  int lane = threadIdx.x;                // one wave32 per 16x16 tile
  int half = lane >> 4;                  // 0: K even pair / M 0..7, 1: K odd pair / M 8..15
  int l16 = lane & 15;
  int b = blockIdx.z;
  int q0 = blockIdx.y * 16;
  int t0 = blockIdx.x * 16;

  const float* A = probs + ((size_t)b * QP + q0 + l16) * CP;  // row m = l16
  const float* B = oh + (size_t)b * CP * TP + t0 + l16;       // col n = l16

  v8f acc = {0.f, 0.f, 0.f, 0.f, 0.f, 0.f, 0.f, 0.f};
#pragma unroll
  for (int kk = 0; kk < CP / 4; ++kk) {
    int kb = 4 * kk + 2 * half;
    v2f a;  a.x = A[kb];                 a.y = A[kb + 1];
    v2f bb; bb.x = B[(size_t)kb * TP];   bb.y = B[(size_t)(kb + 1) * TP];
    acc = __builtin_amdgcn_wmma_f32_16x16x4_f32(false, a, false, bb,
                                                (short)0, acc, false, false);
  }
  // acc[r] = prob[m = r + 8*half][tgt_label[n = l16]]  (exact: one-hot in f32)

  int tn = t0 + l16;
  float tcx = 0.f, tcy = 0.f, tw = 0.f, th = 0.f;
  if (tn < T) {
    const float* tb = tboxes + ((size_t)b * T + tn) * 4;
    tcx = tb[0]; tcy = tb[1]; tw = tb[2]; th = tb[3];
  }
  float tx0 = tcx - 0.5f * tw, ty0 = tcy - 0.5f * th;
  float tx1 = tcx + 0.5f * tw, ty1 = tcy + 0.5f * th;
  float areat = tw * th;

#pragma unroll
  for (int r = 0; r < 8; ++r) {
    int qm = q0 + r + 8 * half;
    if (qm < Q && tn < T) {
      const float* pb = pboxes + ((size_t)b * Q + qm) * 4;
      float pcx = pb[0], pcy = pb[1], pw = pb[2], ph = pb[3];
      float l1 = fabsf(pcx - tcx) + fabsf(pcy - tcy) + fabsf(pw - tw) + fabsf(ph - th);
      float px0 = pcx - 0.5f * pw, py0 = pcy - 0.5f * ph;
      float px1 = pcx + 0.5f * pw, py1 = pcy + 0.5f * ph;
      float areap = pw * ph;
      float ix0 = fmaxf(px0, tx0), iy0 = fmaxf(py0, ty0);
      float ix1 = fminf(px1, tx1), iy1 = fminf(py1, ty1);
      float iw = fmaxf(ix1 - ix0, 0.f), ih = fmaxf(iy1 - iy0, 0.f);
      float inter = iw * ih;
      float uni = areap + areat - inter;
      float iou = inter / uni;
      float cx0 = fminf(px0, tx0), cy0 = fminf(py0, ty0);
      float cx1 = fmaxf(px1, tx1), cy1 = fmaxf(py1, ty1);
      float cw = fmaxf(cx1 - cx0, 0.f), ch = fmaxf(cy1 - cy0, 0.f);
      float areac = cw * ch;
      float giou = iou - (areac - uni) / areac;
      cost[((size_t)b * Q + qm) * TP + tn] = 5.f * l1 - acc[r] - 2.f * giou;
    }
  }
}

// ---------------- anti-diagonal wavefront DP (one WGP per batch) ----------
__global__ void __launch_bounds__(320)
lcs_dp_kernel(const float* __restrict__ cost, float* __restrict__ out) {
  int b = blockIdx.x, tid = threadIdx.x;                 // 320 threads
  float* sc = out + (size_t)b * SCOL * SROW;             // scores plane
  float* pt = out + (size_t)BS * SCOL * SROW + (size_t)b * SCOL * SROW;
  const float* cb = cost + (size_t)b * Q * TP;
  __shared__ float dbuf[3][SCOL];

  for (int i = tid; i < SCOL; i += 320) {
    dbuf[0][i] = 0.f; dbuf[1][i] = 0.f; dbuf[2][i] = 0.f;
  }
  for (int i = tid; i < SCOL; i += 320) {                // boundary col 0
    sc[(size_t)i * SROW] = 0.f;
    pt[(size_t)i * SROW] = (i == 0) ? 0.f : -1.f;
  }
  for (int j = tid; j < SROW; j += 320) {                // boundary row 0
    sc[j] = 0.f;
    pt[j] = (j == 0) ? 0.f : 1.f;
  }
  __syncthreads();

  int cur = 2, p1i = 1, p2i = 0;
  for (int k = 2; k <= Q + T; ++k) {
    float* dk = dbuf[cur];
    const float* d1 = dbuf[p1i];
    const float* d2 = dbuf[p2i];
    int iL = k - (T + 1); if (iL < 0) iL = 0;            // 302-wide sliding window
    int i = iL + tid;
    if (tid < T + 2 && i <= Q) {
      int j = k - i;
      float v = 0.f;
      if (i >= 1 && j >= 1 && j <= T) {
        float rw = 10000.f - cb[(size_t)(i - 1) * TP + (j - 1)];
        float dg = d2[i - 1] + rw;
        float up = d1[i - 1];
        float lf = d1[i];
        float best = fmaxf(dg, fmaxf(up, lf));
        v = best;
        sc[(size_t)i * SROW + j] = best;
        pt[(size_t)i * SROW + j] = (dg >= best) ? 0.f : ((up >= best) ? -1.f : 1.f);
      }
      dk[i] = v;
    }
    __syncthreads();
    int t = p2i; p2i = p1i; p1i = cur; cur = t;
  }
}

// ---------------- traceback (one lane per batch) --------------------------
__global__ void lcs_traceback_kernel(float* __restrict__ out) {
  int b = threadIdx.x;
  if (b >= BS) return;
  const float* pt = out + (size_t)BS * SCOL * SROW + (size_t)b * SCOL * SROW;
  float* mt = out + (size_t)2 * BS * SCOL * SROW + (size_t)b * (Q + T) * 2;
  int r = Q, c = T;
  for (int t = 0; t < Q + T; ++t) {
    float p = pt[(size_t)r * SROW + c];
    bool origin = (r == 0 && c == 0);
    int nr = origin ? 0 : ((p == 1.f) ? r : r - 1);
    int nc = origin ? 0 : ((p == -1.f) ? c : c - 1);
    bool match = (p == 0.f) && !origin;
    size_t o = (size_t)(Q + T - 1 - t) * 2;
    mt[o]     = match ? (float)nr : -1.f;
    mt[o + 1] = match ? (float)nc : -1.f;
    r = nr; c = nc;
  }
}

extern "C" void kernel_launch(void* const* d_in, const int* in_sizes, int n_in,
                              void* d_out, int out_size, void* d_ws, size_t ws_size,
                              hipStream_t stream) {
  const float* logits = (const float*)d_in[0];   // [8,1000,92]
  const float* pboxes = (const float*)d_in[1];   // [8,1000,4]
  const int*   labels = (const int*)d_in[2];     // [8,300]
  const float* tboxes = (const float*)d_in[3];   // [8,300,4]
  float* ws = (float*)d_ws;
  float* probs = ws;                                      // BS*QP*CP
  float* oh    = probs + (size_t)BS * QP * CP;            // BS*CP*TP
  float* cost  = oh + (size_t)BS * CP * TP;               // BS*Q*TP
  float* out   = (float*)d_out;

  lcs_softmax_kernel<<<BS * QP, 128, 0, stream>>>(logits, probs);
  int ohN = BS * CP * TP;
  lcs_onehot_kernel<<<(ohN + 255) / 256, 256, 0, stream>>>(labels, oh);
  dim3 gcost(TP / 16, QP / 16, BS);
  lcs_cost_wmma_kernel<<<gcost, 32, 0, stream>>>(probs, oh, pboxes, tboxes, cost);
  lcs_dp_kernel<<<BS, 320, 0, stream>>>(cost, out);
  lcs_traceback_kernel<<<1, 32, 0, stream>>>(out);
}